// VoxelSampler_4123168604647
// MI455X (gfx1250) — compile-verified
//
#include <hip/hip_runtime.h>

typedef __attribute__((ext_vector_type(2))) float v2f;
typedef __attribute__((ext_vector_type(8))) float v8f;
typedef __attribute__((ext_vector_type(4))) unsigned int u32x4;
typedef __attribute__((ext_vector_type(8))) int i32x8;
typedef __attribute__((ext_vector_type(4))) int i32x4;

namespace {
constexpr int   T_FRAMES = 8;
constexpr int   R_ROIS   = 128;
constexpr int   N_PTS    = 65536;
constexpr int   C_FEAT   = 5;
constexpr int   K_SAMPLE = 32;
constexpr int   TILE     = 16;             // rois per workgroup (one WMMA M-tile)
constexpr int   CHUNK    = 64;             // points per TDM-staged chunk (4 WMMA N-tiles)
constexpr int   CHUNK_F  = CHUNK * C_FEAT; // floats per staged chunk (320)
constexpr int   NCHUNK   = N_PTS / CHUNK;  // 1024
constexpr float GAMMA_C  = 1.05f;
}

// Issue a TDM copy of CHUNK_F f32 from global to LDS (1-D tile descriptor).
// D# packing per CDNA5 ISA ch.8 (group0: count/lds_addr/global_addr/type,
// group1: data_size=4B, tensor_dim0=tile_dim0=CHUNK_F, dim1=1, groups 2/3 zero).
// This toolchain exposes the 6-arg builtin: (g0, g1, g2, g3, g4, cpol).
__device__ __forceinline__ void tdm_load_chunk(const float* gsrc, unsigned lds_off) {
  unsigned long long ga = (unsigned long long)(size_t)gsrc;
  u32x4 g0;
  g0[0] = 1u;                                     // count=1 (valid user descriptor)
  g0[1] = lds_off;                                // lds_addr (bytes)
  g0[2] = (unsigned)(ga & 0xFFFFFFFFull);         // global_addr[31:0]
  g0[3] = (unsigned)((ga >> 32) & 0x01FFFFFFull)  // global_addr[56:32]
        | (2u << 30);                             // type=2 ("image")
  i32x8 g1;
  g1[0] = (2 << 16);                              // wg_mask=0 | data_size=2 (4 bytes)
  g1[1] = (int)((CHUNK_F & 0xFFFF) << 16);        // tensor_dim0[15:0]   (bits 63:48)
  g1[2] = (int)((CHUNK_F >> 16) & 0xFFFF)         // tensor_dim0[31:16]  (bits 79:64)
        | (1 << 16);                              // tensor_dim1[15:0]=1 (bits 95:80)
  g1[3] = (int)(CHUNK_F << 16);                   // tile_dim0 = 320     (bits 127:112)
  g1[4] = 1;                                      // tile_dim1=1, tile_dim2=0
  g1[5] = CHUNK_F;                                // tensor_dim0_stride[31:0]
  g1[6] = 0;                                      // stride0 hi / stride1 lo
  g1[7] = 0;
  i32x4 gz4 = {0, 0, 0, 0};                       // groups 2/3: dims unused
  i32x8 gz8 = {0, 0, 0, 0, 0, 0, 0, 0};           // extra group (clang-23 form): zero
  __builtin_amdgcn_tensor_load_to_lds(g0, g1, gz4, gz4, gz8, 0);
}

// One wave32 per (frame, 16-roi tile).
// d^2(r,n) = |c|^2 - 2 c.p + |p|^2  ==  A(16x4) x B(4x16) via V_WMMA_F32_16X16X4_F32
//   A row r = [cx, cy, |c|^2, 1]     B col n = [-2px, -2py, 1, |p|^2]
// Point data is double-buffered into LDS by the Tensor Data Mover (TENSORcnt),
// then wave32 ballot/popcount performs the stable "first 32 in index order"
// compaction that jax.lax.top_k over a 0/1 mask implements.
__global__ __launch_bounds__(32)
void voxel_sampler_wmma(const float* __restrict__ points,  // [T, N, 5]
                        const float* __restrict__ boxes,   // [T, R, 9]
                        float* __restrict__ out)           // [R, T, 32, 5]
{
  const int bid    = blockIdx.x;
  const int t      = bid >> 3;            // frame
  const int tb     = (bid & 7) * TILE;    // roi tile base
  const int lane   = threadIdx.x;
  const int half   = (lane >= 16) ? 8 : 0;
  const int bitpos = lane & 15;           // point slot within a 16-wide sub-tile

  __shared__ float stage[2][CHUNK_F];     // TDM double buffer (2 x 1280 B)
  __shared__ float rad2_s[TILE];
  __shared__ float cx_s[TILE], cy_s[TILE], c2_s[TILE];
  __shared__ int   cnt[TILE];

  // Per-roi setup (radius^2, center terms) by lanes 0..15.
  if (lane < TILE) {
    const float* bx = boxes + ((size_t)t * R_ROIS + (tb + lane)) * 9;
    float cx = bx[0], cy = bx[1];
    float dx = bx[3], dy = bx[4];
    float vx = bx[7], vy = bx[8];
    float speed = sqrtf(vx * vx + vy * vy);
    float g = powf(GAMMA_C * (1.0f + speed), (float)t * 0.2f);  // t/5
    g = fminf(g, 2.0f);
    float rad = 0.5f * sqrtf(dx * dx + dy * dy) * g;
    rad2_s[lane] = rad * rad;
    cx_s[lane] = cx;
    cy_s[lane] = cy;
    c2_s[lane] = cx * cx + cy * cy;
    cnt[lane]  = 0;
  }
  __syncthreads();

  // Zero this tile's output region out[tb..tb+15, t, :, :] (pad semantics +
  // deterministic across poisoned replays).
  for (int i = lane; i < TILE * K_SAMPLE * C_FEAT; i += 32) {
    int r   = i / (K_SAMPLE * C_FEAT);
    int rem = i - r * (K_SAMPLE * C_FEAT);
    out[(((size_t)(tb + r) * T_FRAMES + t) * K_SAMPLE) * C_FEAT + rem] = 0.0f;
  }

  // A operand: 16x4 f32. Lanes 0-15 carry K=0,1 (cx,cy); lanes 16-31 K=2,3 (|c|^2, 1).
  v2f A;
  if (lane < 16) { A.x = cx_s[lane];      A.y = cy_s[lane]; }
  else           { A.x = c2_s[lane - 16]; A.y = 1.0f; }

  // Per-lane radius^2 for each C VGPR slot v: roi = v + half.
  float rad2v[8];
#pragma unroll
  for (int v = 0; v < 8; ++v) rad2v[v] = rad2_s[v + half];

  const float* ptsT = points + (size_t)t * N_PTS * C_FEAT;
  const unsigned lds0 = (unsigned)(size_t)(void*)&stage[0][0];
  const unsigned lds1 = (unsigned)(size_t)(void*)&stage[1][0];

  // Prologue: stage chunk 0.
  tdm_load_chunk(ptsT, lds0);

  for (int c = 0; c < NCHUNK; ++c) {
    // Uniform early exit once all 16 rois have 32 samples.
    if (__all(cnt[bitpos] >= K_SAMPLE)) break;

    const int cb = c & 1;
    if (c + 1 < NCHUNK) {
      tdm_load_chunk(ptsT + (size_t)(c + 1) * CHUNK_F, cb ? lds0 : lds1);
      __builtin_amdgcn_s_wait_tensorcnt(1);  // chunk c landed; c+1 in flight
    } else {
      __builtin_amdgcn_s_wait_tensorcnt(0);
    }
    asm volatile("" ::: "memory");  // keep LDS reads below the tensorcnt wait

    const float* sbuf = &stage[cb][0];

#pragma unroll
    for (int s = 0; s < 4; ++s) {
      const int pt = s * 16 + bitpos;          // point index within chunk
      float px = sbuf[pt * C_FEAT + 0];
      float py = sbuf[pt * C_FEAT + 1];

      // B operand: 4x16 f32. Lanes 0-15: K=0,1 (-2px,-2py); lanes 16-31: K=2,3 (1,|p|^2).
      v2f B;
      if (lane < 16) { B.x = -2.0f * px; B.y = -2.0f * py; }
      else           { B.x = 1.0f;       B.y = px * px + py * py; }

      v8f d2 = {0.f, 0.f, 0.f, 0.f, 0.f, 0.f, 0.f, 0.f};
      d2 = __builtin_amdgcn_wmma_f32_16x16x4_f32(
          /*neg_a=*/false, A, /*neg_b=*/false, B,
          /*c_mod=*/(short)0, d2, /*reuse_a=*/false, /*reuse_b=*/false);

      // C/D layout: VGPR v, lane l -> roi (v + 8*(l>=16)), point (s*16 + l%16).
      unsigned bal[8];
      unsigned any = 0;
#pragma unroll
      for (int v = 0; v < 8; ++v) {
        bal[v] = (unsigned)__ballot(d2[v] <= rad2v[v]);
        any |= bal[v];
      }
      if (any == 0) continue;  // uniform: whole 16x16 tile empty

#pragma unroll
      for (int v = 0; v < 8; ++v) {
        unsigned b = bal[v];
        if (b == 0) continue;  // uniform skip
        unsigned mm = (lane < 16) ? (b & 0xFFFFu) : (b >> 16);
        int roi   = v + half;
        int basec = cnt[roi];
        if ((mm >> bitpos) & 1u) {
          int rank = __popc(mm & ((1u << bitpos) - 1u));
          int slot = basec + rank;
          if (slot < K_SAMPLE) {
            const float* src = sbuf + pt * C_FEAT;  // features from LDS stage
            float* dst = out + (((size_t)(tb + roi) * T_FRAMES + t) * K_SAMPLE + slot) * C_FEAT;
            dst[0] = src[0]; dst[1] = src[1]; dst[2] = src[2];
            dst[3] = src[3]; dst[4] = src[4];
          }
        }
        if (bitpos == 0) cnt[roi] = basec + __popc(mm);  // lanes 0/16: disjoint rois
      }
    }
  }
}

extern "C" void kernel_launch(void* const* d_in, const int* in_sizes, int n_in,
                              void* d_out, int out_size, void* d_ws, size_t ws_size,
                              hipStream_t stream) {
  (void)in_sizes; (void)n_in; (void)d_ws; (void)ws_size; (void)out_size;
  const float* points = (const float*)d_in[0];   // [8, 65536, 5] f32
  const float* boxes  = (const float*)d_in[1];   // [8, 128, 9]   f32
  // d_in[2] = num_sample (always 32 per harness; K hardcoded)
  float* out = (float*)d_out;                    // [128, 8, 32, 5] f32

  dim3 grid(T_FRAMES * (R_ROIS / TILE));  // 64 workgroups
  dim3 block(32);                         // one wave32 each
  voxel_sampler_wmma<<<grid, block, 0, stream>>>(points, boxes, out);
}